// EnhancedLIFNeuron_79860621902351
// MI455X (gfx1250) — compile-verified
//
#include <hip/hip_runtime.h>
#include <hip/hip_bf16.h>
#include <math.h>

// ---- MI455X / gfx1250 LIF kernel -------------------------------------------
// x[B,T,F] --TDM--> LDS tile [T x 64f] --WMMA banded conv--> LDS y tile
// --> per-(b,f) attention + LIF scan in registers --> coalesced spike stores.
//
// Hot conv loop is branchless: A-fragments (band coefficients) are precomputed
// once per wave (tile-invariant), boundary K-chunks are zeroed via wave-uniform
// selects on the A side, B loads are unconditional ds_load_b32.

typedef float v2f  __attribute__((ext_vector_type(2)));
typedef float v8f  __attribute__((ext_vector_type(8)));
typedef unsigned int u32x4 __attribute__((ext_vector_type(4)));
typedef int   i32x4 __attribute__((ext_vector_type(4)));
typedef int   i32x8 __attribute__((ext_vector_type(8)));

#define B_  128
#define T_  128
#define F_  2048
#define FC  64      // features per block
#define NT  64      // threads per block (2 waves of 32)

__global__ __launch_bounds__(NT) void lif_wmma_tdm_kernel(
    const float* __restrict__ x,
    const float* __restrict__ init_threshold,
    const float* __restrict__ conv_w,
    const float* __restrict__ conv_b,
    const float* __restrict__ spatial_attn,
    float* __restrict__ out)
{
    __shared__ float lds_x[T_ * FC];   // raw x tile  [t][f_local]  32 KB
    __shared__ float lds_y[T_ * FC];   // conv output [t][f_local]  32 KB

    const int tid = threadIdx.x;
    const int b   = blockIdx.x >> 5;          // F_/FC = 32 feature chunks
    const int f0  = (blockIdx.x & 31) * FC;

    // ---------- Phase 0: TDM async stage of x tile into LDS ----------
    if ((tid >> 5) == 0) {                    // one wave issues the DMA
#if defined(__gfx1250__)
        unsigned long long ga =
            (unsigned long long)(const void*)x +
            ((unsigned long long)b * T_ * F_ + (unsigned long long)f0) * 4ull;
        unsigned int lds_off =
            (unsigned int)(unsigned long long)(const void*)&lds_x[0];

        u32x4 g0;
        g0.x = 1u;                                      // count = 1 (valid D#)
        g0.y = lds_off;                                 // lds_addr
        g0.z = (unsigned int)ga;                        // global_addr[31:0]
        g0.w = ((unsigned int)(ga >> 32) & 0x01FFFFFFu) // global_addr[56:32]
             | (2u << 30);                              // type = 2 ("image")

        i32x8 g1;
        g1[0] = (int)(2u << 16);                 // data_size = 2 -> 4 bytes
        g1[1] = (int)((unsigned)F_ << 16);       // tensor_dim0[15:0]  (2048)
        g1[2] = (int)((unsigned)T_ << 16);       // tensor_dim1[15:0]  (128)
        g1[3] = (int)((unsigned)FC << 16);       // tile_dim0          (64)
        g1[4] = (int)T_;                         // tile_dim1          (128)
        g1[5] = (int)F_;                         // tensor_dim0_stride (2048)
        g1[6] = 0;
        g1[7] = 0;
        i32x4 g2 = {0, 0, 0, 0};
        i32x4 g3 = {0, 0, 0, 0};
#if __clang_major__ >= 23
        i32x8 g4 = {0, 0, 0, 0, 0, 0, 0, 0};
        __builtin_amdgcn_tensor_load_to_lds(g0, g1, g2, g3, g4, 0);
#else
        __builtin_amdgcn_tensor_load_to_lds(g0, g1, g2, g3, 0);
#endif
        __builtin_amdgcn_s_wait_tensorcnt(0);
#else
        for (int i = tid; i < T_ * FC; i += 32) {
            int t = i / FC, f = i % FC;
            lds_x[i] = x[((size_t)b * T_ + t) * F_ + (f0 + f)];
        }
#endif
    }
    __syncthreads();

    // Scalar parameters (wave-uniform loads through constant/L2 path)
    const float thr0 = init_threshold[0];
    const float w0 = conv_w[0], w1 = conv_w[1], w2 = conv_w[2],
                w3 = conv_w[3], w4 = conv_w[4];
    const float cb = conv_b[0];
    const float sa = spatial_attn[0];

    // ---------- Phase 1: banded-GEMM temporal conv via WMMA ----------
    // Y[t,f] = sum_k C[t,k] * X[k,f],  C[t,k] = w[k - t + 2] on the band.
    const int lane  = tid & 31;
    const int wave  = tid >> 5;
    const int mn    = lane & 15;   // A: row M; B: col N; D: col N
    const int khalf = lane >> 4;   // selects K pair (A/B) / row half (D)

    // A-fragments are the same for every tile: idx = 4j + kk - mn - 2.
    // Computed once per wave, kept in registers (constant indices via unroll).
    v2f afrag[6];
    #pragma unroll
    for (int j = 0; j < 6; ++j) {
        #pragma unroll
        for (int v = 0; v < 2; ++v) {
            const int kk  = (khalf << 1) | v;
            const int idx = 4 * j + kk - mn - 2;
            afrag[j][v] = (idx == 0) ? w0 : (idx == 1) ? w1 : (idx == 2) ? w2
                        : (idx == 3) ? w3 : (idx == 4) ? w4 : 0.0f;
        }
    }

    for (int tile = wave; tile < (T_ / 16) * (FC / 16); tile += (NT / 32)) {
        const int t0  = (tile >> 2) << 4;   // 8 row tiles (time), wave-uniform
        const int f0l = (tile & 3) << 4;    // 4 col tiles (features)
        const int col = f0l + mn;
        v8f acc = {};
#if defined(__gfx1250__)
        #pragma unroll
        for (int j = 0; j < 6; ++j) {       // K-chunks covering band [t0-4, t0+20)
            v2f a = afrag[j];
            // Boundary chunks are entirely out of range; zero A under a
            // wave-uniform condition (t0 uniform) -- no per-lane predication.
            if (j == 0) { if (t0 == 0)        { a[0] = 0.0f; a[1] = 0.0f; } }
            if (j == 5) { if (t0 == T_ - 16)  { a[0] = 0.0f; a[1] = 0.0f; } }
            int k0 = t0 - 4 + 4 * j + (khalf << 1);
            k0 = (k0 < 0) ? 0 : k0;          // clamp: A==0 nullifies clamped rows
            k0 = (k0 > T_ - 2) ? (T_ - 2) : k0;
            v2f bm;
            bm[0] = lds_x[k0 * FC + col];            // unconditional ds_load
            bm[1] = lds_x[(k0 + 1) * FC + col];
            acc = __builtin_amdgcn_wmma_f32_16x16x4_f32(
                      false, a, false, bm, (short)0, acc, false, false);
        }
#else
        // Scalar-equivalent fallback (host pass / non-gfx1250; never runs on device)
        const float wv[5] = {w0, w1, w2, w3, w4};
        for (int r = 0; r < 8; ++r) {
            const int t = t0 + (khalf << 3) + r;
            float s = 0.0f;
            for (int kk2 = 0; kk2 < 5; ++kk2) {
                const int k = t + kk2 - 2;
                if (k >= 0 && k < T_) s += wv[kk2] * lds_x[k * FC + f0l + mn];
            }
            acc[r] = s;
        }
#endif
        // D 16x16 f32 layout: VGPR r, lane-half khalf -> row t0 + 8*khalf + r
        #pragma unroll
        for (int r = 0; r < 8; ++r)
            lds_y[(t0 + (khalf << 3) + r) * FC + f0l + mn] = acc[r];
    }
    __syncthreads();

    // ---------- Phase 2: attention scale + LIF recurrence ----------
    const int f = f0 + tid;      // one (b,f) per thread

    float sum = 0.0f;
    {
        const float* p = &lds_y[tid];
        for (int t = 0; t < T_; ++t, p += FC) sum += *p;
    }
    const float mean  = sum * (1.0f / (float)T_) + cb;
    const float attw  = 1.0f / (1.0f + expf(-sa * mean));   // sigmoid
    const float scale = 1.0f + 0.5f * attw;                 // GAMMA = 0.5

    float memv = 0.0f, thr = thr0;
    float h0 = 0.0f, h1 = 0.0f, h2 = 0.0f;
    const float PI     = 3.14159265358979323846f;
    const float INV_PI = 0.31830988618379067154f;

    float* spk = out + ((size_t)b * T_) * F_ + f;   // steps by F_ per t
    const float* py = &lds_y[tid];
    for (int t = 0; t < T_; ++t, spk += F_, py += FC) {
        const float xt = (*py + cb) * scale;
        memv = 0.9f * memv + xt;                                     // DECAY
        const float spike =
            0.5f * (atanf(PI * (memv - thr) * 0.5f) * INV_PI + 0.5f);
        h0 = h1; h1 = h2; h2 = spike;                                // hist window
        const float avg  = (h0 + h1 + h2) * (1.0f / 3.0f);
        const float nthr = 0.9f * thr + 0.1f * thr0 + 0.1f * avg;    // ALPHA/BETA
        memv = memv - spike * nthr;
        thr  = nthr;
        *spk = spike;                                                // coalesced
    }
    out[(size_t)B_ * T_ * F_ + (size_t)b * F_ + f] = memv;           // mem tail
}

extern "C" void kernel_launch(void* const* d_in, const int* in_sizes, int n_in,
                              void* d_out, int out_size, void* d_ws, size_t ws_size,
                              hipStream_t stream) {
    const float* x   = (const float*)d_in[0];
    const float* it  = (const float*)d_in[1];
    const float* cw  = (const float*)d_in[2];
    const float* cbp = (const float*)d_in[3];
    const float* sap = (const float*)d_in[4];
    float* out = (float*)d_out;

    dim3 grid(B_ * (F_ / FC));   // 4096 blocks
    dim3 block(NT);              // 64 threads = 2 waves
    lif_wmma_tdm_kernel<<<grid, block, 0, stream>>>(x, it, cw, cbp, sap, out);

    (void)in_sizes; (void)n_in; (void)out_size; (void)d_ws; (void)ws_size;
}